// GCN_16879221473613
// MI455X (gfx1250) — compile-verified
//
#include <hip/hip_runtime.h>
#include <hip/hip_bf16.h>

typedef float v2f __attribute__((ext_vector_type(2)));
typedef float v8f __attribute__((ext_vector_type(8)));

#define BN_EPS 1e-5f

// ---------------- utility: fill ----------------
__global__ void k_fill(float* __restrict__ p, float v, long long count) {
    long long i = blockIdx.x * (long long)blockDim.x + threadIdx.x;
    if (i < count) p[i] = v;
}

// ---------------- degree count (dst side), deg pre-initialized to 1.0 (self loop)
__global__ void k_count_deg(const int* __restrict__ dst, float* __restrict__ deg, int E) {
    int e = blockIdx.x * blockDim.x + threadIdx.x;
    if (e < E) atomicAdd(&deg[dst[e]], 1.0f);
}

__global__ void k_make_dinv(float* __restrict__ d, int n) {
    int i = blockIdx.x * blockDim.x + threadIdx.x;
    if (i < n) {
        float v = d[i];
        d[i] = (v > 0.0f) ? rsqrtf(v) : 0.0f;
    }
}

// ---------------- fp32 WMMA GEMM: C[n x 32] = X[n x K] @ W[K x 32] ----------------
// One wave per 16x16 output tile. 2 column tiles (out width 32).
// A layout (16x4 f32, ISA 7.12.2): lanes 0-15 -> M=lane, VGPR0=K+0, VGPR1=K+1;
//                                  lanes 16-31 -> M=lane-16, VGPR0=K+2, VGPR1=K+3.
// B layout mirrors A with N in place of M. C/D: N=lane%16, M = r + 8*(lane/16).
// Rows are CLAMPED (not predicated) so all loads are unconditional -> no exec
// churn; padded rows compute garbage that the store guard discards.
template <int K>
__global__ void k_gemm_wmma(const float* __restrict__ X, const float* __restrict__ W,
                            float* __restrict__ C, int n) {
    int gid  = blockIdx.x * blockDim.x + threadIdx.x;
    int wave = gid >> 5;
    int lane = gid & 31;
    int row0 = (wave >> 1) << 4;
    int col0 = (wave & 1) << 4;
    if (row0 >= n) return;

    int half = lane >> 4;      // 0: lanes 0-15, 1: lanes 16-31
    int l    = lane & 15;
    int arow = row0 + l;
    if (arow > n - 1) arow = n - 1;              // clamp: keep loads unconditional
    const float* __restrict__ xr = X + (long long)arow * K + (half << 1);
    const float* __restrict__ wc = W + (half << 1) * 32 + (col0 + l);

    v8f c = {0.f, 0.f, 0.f, 0.f, 0.f, 0.f, 0.f, 0.f};
#pragma unroll
    for (int k = 0; k < K; k += 4) {
        v2f a = *reinterpret_cast<const v2f*>(xr + k);   // K even, 8B aligned
        v2f b;
        b.x = wc[k * 32];
        b.y = wc[k * 32 + 32];
        c = __builtin_amdgcn_wmma_f32_16x16x4_f32(false, a, false, b,
                                                  (short)0, c, false, false);
    }
#pragma unroll
    for (int r = 0; r < 8; ++r) {
        int m = row0 + r + (half << 3);
        if (m < n) C[(long long)m * 32 + col0 + l] = c[r];
    }
}

// ---------------- edge scatter: out[dst] += dinv[s]*dinv[d] * h[src] ----------------
// One wave per edge (incl. N appended self-loops); lane = channel. Coalesced 128B
// gather + 128B atomic burst per edge; h fits in L2 (25.6MB << 192MB).
__global__ void k_scatter(const float* __restrict__ h, const int* __restrict__ src,
                          const int* __restrict__ dst, const float* __restrict__ dinv,
                          float* __restrict__ out, long long E, long long Et) {
    long long t = blockIdx.x * (long long)blockDim.x + threadIdx.x;
    long long e = t >> 5;
    if (e >= Et) return;
    int c = (int)(t & 31);
    int s, d;
    if (e < E) { s = src[e]; d = dst[e]; }
    else       { s = d = (int)(e - E); }
    float nrm = dinv[s] * dinv[d];
    atomicAdd(&out[(long long)d * 32 + c], nrm * h[(long long)s * 32 + c]);
}

// ---------------- bias + tanh (+ per-channel sum/sumsq stats) ----------------
// blockDim must be 256. Grid-stride (stride % 32 == 0 keeps channel stable).
__global__ void k_post(float* __restrict__ h, const float* __restrict__ bias,
                       float* __restrict__ stats, long long total, int do_stats) {
    int c = threadIdx.x & 31;
    float b = bias[c];
    float ls = 0.f, lq = 0.f;
    long long stride = (long long)gridDim.x * blockDim.x;
    for (long long i = blockIdx.x * (long long)blockDim.x + threadIdx.x;
         i < total; i += stride) {
        float t = tanhf(h[i] + b);
        h[i] = t;
        ls += t;
        lq += t * t;
    }
    if (do_stats) {
        __shared__ float ss[256];
        __shared__ float sq[256];
        ss[threadIdx.x] = ls;
        sq[threadIdx.x] = lq;
        __syncthreads();
        if (threadIdx.x < 32) {
            float a = 0.f, q = 0.f;
#pragma unroll
            for (int g = 0; g < 8; ++g) {
                a += ss[g * 32 + threadIdx.x];
                q += sq[g * 32 + threadIdx.x];
            }
            atomicAdd(&stats[threadIdx.x], a);
            atomicAdd(&stats[32 + threadIdx.x], q);
        }
    }
}

// ---------------- batchnorm apply (train-mode, biased var) ----------------
__global__ void k_bn(float* __restrict__ h, const float* __restrict__ stats,
                     const float* __restrict__ g, const float* __restrict__ be,
                     long long total, float invN) {
    long long stride = (long long)gridDim.x * blockDim.x;
    for (long long i = blockIdx.x * (long long)blockDim.x + threadIdx.x;
         i < total; i += stride) {
        int c = (int)(i & 31);
        float mu  = stats[c] * invN;
        float var = stats[32 + c] * invN - mu * mu;
        h[i] = (h[i] - mu) * rsqrtf(var + BN_EPS) * g[c] + be[c];
    }
}

// ---------------- global mean pool: accumulate sums + counts ----------------
__global__ void k_pool(const float* __restrict__ h, const int* __restrict__ batch,
                       float* __restrict__ pool, float* __restrict__ cnt, long long n) {
    long long t = blockIdx.x * (long long)blockDim.x + threadIdx.x;
    long long node = t >> 5;
    if (node >= n) return;
    int c = (int)(t & 31);
    int g = batch[node];
    atomicAdd(&pool[(long long)g * 32 + c], h[node * 32 + c]);
    if (c == 0) atomicAdd(&cnt[g], 1.0f);
}

// ---------------- classifier: out[g] = mean(h_g) . Wc + bc ----------------
__global__ void k_cls(const float* __restrict__ pool, const float* __restrict__ cnt,
                      const float* __restrict__ Wc, const float* __restrict__ bc,
                      float* __restrict__ out, int G) {
    int gid  = blockIdx.x * blockDim.x + threadIdx.x;
    int wave = gid >> 5;
    int lane = gid & 31;
    if (wave >= G) return;
    float v = pool[(long long)wave * 32 + lane] / fmaxf(cnt[wave], 1.0f);
    float p = v * Wc[lane];
#pragma unroll
    for (int off = 16; off > 0; off >>= 1) p += __shfl_xor(p, off, 32);
    if (lane == 0) out[wave] = p + bc[0];
}

extern "C" void kernel_launch(void* const* d_in, const int* in_sizes, int n_in,
                              void* d_out, int out_size, void* d_ws, size_t ws_size,
                              hipStream_t stream) {
    const float* x    = (const float*)d_in[0];
    const int*   eidx = (const int*)d_in[1];
    const int*   batch= (const int*)d_in[2];
    const float* W1 = (const float*)d_in[3];
    const float* b1 = (const float*)d_in[4];
    const float* g1 = (const float*)d_in[5];
    const float* be1= (const float*)d_in[6];
    const float* W2 = (const float*)d_in[7];
    const float* b2 = (const float*)d_in[8];
    const float* g2 = (const float*)d_in[9];
    const float* be2= (const float*)d_in[10];
    const float* W3 = (const float*)d_in[11];
    const float* b3 = (const float*)d_in[12];
    const float* Wc = (const float*)d_in[15];
    const float* bc = (const float*)d_in[16];
    float* out = (float*)d_out;

    long long n = in_sizes[0] / 128;
    long long E = in_sizes[1] / 2;
    int G = out_size;
    const int* src = eidx;
    const int* dst = eidx + E;

    // workspace carve-out (256B aligned): dinv | hA | hB | stats | pool+cnt
    char* ws = (char*)d_ws;
    size_t off = 0;
    auto carve = [&](size_t bytes) -> float* {
        float* p = (float*)(ws + off);
        off = (off + bytes + 255) & ~(size_t)255;
        return p;
    };
    float* dinv  = carve((size_t)n * 4);
    float* hA    = carve((size_t)n * 32 * 4);
    float* hB    = carve((size_t)n * 32 * 4);
    float* stats = carve(64 * 4);
    float* pool  = carve((size_t)G * 33 * 4);   // G*32 sums then G counts
    float* cnt   = pool + (size_t)G * 32;

    const int B = 256;
    auto blocks = [&](long long t) -> unsigned { return (unsigned)((t + B - 1) / B); };

    long long total    = n * 32;
    long long Et       = E + n;
    long long scatterT = Et * 32;
    long long gemmT    = ((n + 15) / 16) * 2 * 32;   // waves * 32 threads
    float invN = 1.0f / (float)n;

    // --- normalization: deg (init 1.0 for self-loop) -> dinv ---
    k_fill<<<blocks(n), B, 0, stream>>>(dinv, 1.0f, n);
    k_count_deg<<<blocks(E), B, 0, stream>>>(dst, dinv, (int)E);
    k_make_dinv<<<blocks(n), B, 0, stream>>>(dinv, (int)n);

    // --- layer 1: 128 -> 32, tanh, BN ---
    k_gemm_wmma<128><<<blocks(gemmT), B, 0, stream>>>(x, W1, hA, (int)n);
    k_fill<<<blocks(total), B, 0, stream>>>(hB, 0.0f, total);
    k_scatter<<<blocks(scatterT), B, 0, stream>>>(hA, src, dst, dinv, hB, E, Et);
    k_fill<<<1, 64, 0, stream>>>(stats, 0.0f, 64);
    k_post<<<2048, B, 0, stream>>>(hB, b1, stats, total, 1);
    k_bn<<<2048, B, 0, stream>>>(hB, stats, g1, be1, total, invN);

    // --- layer 2: 32 -> 32, tanh, BN ---
    k_gemm_wmma<32><<<blocks(gemmT), B, 0, stream>>>(hB, W2, hA, (int)n);
    k_fill<<<blocks(total), B, 0, stream>>>(hB, 0.0f, total);
    k_scatter<<<blocks(scatterT), B, 0, stream>>>(hA, src, dst, dinv, hB, E, Et);
    k_fill<<<1, 64, 0, stream>>>(stats, 0.0f, 64);
    k_post<<<2048, B, 0, stream>>>(hB, b2, stats, total, 1);
    k_bn<<<2048, B, 0, stream>>>(hB, stats, g2, be2, total, invN);

    // --- layer 3: 32 -> 32, tanh (no BN) ---
    k_gemm_wmma<32><<<blocks(gemmT), B, 0, stream>>>(hB, W3, hA, (int)n);
    k_fill<<<blocks(total), B, 0, stream>>>(hB, 0.0f, total);
    k_scatter<<<blocks(scatterT), B, 0, stream>>>(hA, src, dst, dinv, hB, E, Et);
    k_post<<<2048, B, 0, stream>>>(hB, b3, stats, total, 0);

    // --- mean pool + classifier ---
    k_fill<<<blocks((long long)G * 33), B, 0, stream>>>(pool, 0.0f, (long long)G * 33);
    k_pool<<<blocks(total), B, 0, stream>>>(hB, batch, pool, cnt, n);
    k_cls<<<blocks((long long)G * 32), B, 0, stream>>>(pool, cnt, Wc, bc, out, G);
}